// SelfAttentionLayer_52853867545143
// MI455X (gfx1250) — compile-verified
//
#include <hip/hip_runtime.h>
#include <math.h>

// Self-attention pooling: scores = inputs @ W_in (+ row-constant terms that
// cancel in softmax), weights = softmax(scores over 1e6 rows),
// out[128] = sum_i weights[i] * inputs[i].
// Single-pass online-softmax streaming kernel (HBM-bound: 512MB once),
// scores computed exactly with V_WMMA_F32_16X16X4_F32.

#define NINS    1000000
#define NHID    128
#define NB      1024            // blocks in pass 1
#define BT      256             // threads per block (8 wave32)
#define WPB     (BT / 32)
#define TOTW    (NB * WPB)      // 8192 waves
#define NTILES  (NINS / 16)     // 62500 tiles of 16 rows (exact)
#define PSTRIDE (NHID + 2)      // per-block partial: [m, s, acc[128]]

typedef float v2f __attribute__((ext_vector_type(2)));
typedef float v8f __attribute__((ext_vector_type(8)));

// wave-uniform broadcast of a lane's float via v_readlane_b32 (no LDS traffic)
__device__ __forceinline__ float uread(float x, int srcLane) {
  return __int_as_float(__builtin_amdgcn_readlane(__float_as_int(x), srcLane));
}

__global__ __launch_bounds__(BT)
void sa_partial(const float* __restrict__ inputs,
                const float* __restrict__ W,
                float* __restrict__ part) {
  __shared__ float w_lds[NHID];
  __shared__ float wacc[WPB][NHID + 4];   // +4 pad vs bank conflicts
  __shared__ float wm[WPB];
  __shared__ float wsum[WPB];

  const int tid = threadIdx.x;
  if (tid < NHID) w_lds[tid] = W[NHID + tid];   // W_in = W[128:256]
  __syncthreads();

  const int lane = tid & 31;
  const int wave = tid >> 5;
  const int half = lane >> 4;     // 0: lanes 0-15 (K pair 0,1) ; 1: lanes 16-31 (K pair 2,3)
  const int mrow = lane & 15;     // A-matrix row index M
  const int gw   = blockIdx.x * WPB + wave;

  float  m = -INFINITY;
  float  s = 0.0f;
  float4 acc = make_float4(0.f, 0.f, 0.f, 0.f);   // lane owns cols 4*lane..4*lane+3

  for (int t = gw; t < NTILES; t += TOTW) {
    const float* tile = inputs + (size_t)t * (16 * NHID);

    // prefetch next tile (8KB = 64 cachelines): global_prefetch_b8
    if (t + TOTW < NTILES) {
      const char* nt = (const char*)(inputs + (size_t)(t + TOTW) * (16 * NHID));
      __builtin_prefetch(nt + lane * 128, 0, 3);
      __builtin_prefetch(nt + 4096 + lane * 128, 0, 3);
    }

    // ---- phase 1: 16 row-scores via V_WMMA_F32_16X16X4_F32 ----
    // A: 16x4 f32 tile rows; lane l holds A[l%16, 2*(l/16)] , A[l%16, 2*(l/16)+1]
    // B: 4x16 with every column = W_in chunk -> all D columns identical
    v8f c = {0.f, 0.f, 0.f, 0.f, 0.f, 0.f, 0.f, 0.f};
    const float* arow = tile  + mrow * NHID + 2 * half;
    const float* brow = w_lds + 2 * half;
#pragma unroll 8
    for (int k0 = 0; k0 < NHID; k0 += 4) {
      v2f a = *(const v2f*)(arow + k0);
      v2f b = *(const v2f*)(brow + k0);
      c = __builtin_amdgcn_wmma_f32_16x16x4_f32(false, a, false, b,
                                                (short)0, c, false, false);
    }
    // D layout: lane l, VGPR v == score(row v + 8*(l/16)); all N columns equal,
    // so lanes 0-15 hold rows 0-7 and lanes 16-31 hold rows 8-15 (replicated).

    // per-lane max over its 8 scores, then 2 readlane broadcasts for tile max
    float vmax = fmaxf(fmaxf(fmaxf(c[0], c[1]), fmaxf(c[2], c[3])),
                       fmaxf(fmaxf(c[4], c[5]), fmaxf(c[6], c[7])));
    const float tmax = fmaxf(uread(vmax, 0), uread(vmax, 16));

    // ---- online softmax update (readlane-uniform -> scalar branch) ----
    if (tmax > m) {
      const float scale = __expf(m - tmax);   // exp(-inf)=0 on first tile
      s *= scale;
      acc.x *= scale; acc.y *= scale; acc.z *= scale; acc.w *= scale;
      m = tmax;
    }

    // ---- fused exp + weighted accumulate; tile re-read hits L0 ----
    const float* col = tile + 4 * lane;
#pragma unroll
    for (int r = 0; r < 16; ++r) {
      const float sc = uread(c[r & 7], (r >> 3) * 16);
      const float e  = __expf(sc - m);
      s += e;
      const float4 x = *(const float4*)(col + r * NHID);
      acc.x += e * x.x;
      acc.y += e * x.y;
      acc.z += e * x.z;
      acc.w += e * x.w;
    }
  }

  // ---- merge 8 waves within the block ----
  wacc[wave][4 * lane + 0] = acc.x;
  wacc[wave][4 * lane + 1] = acc.y;
  wacc[wave][4 * lane + 2] = acc.z;
  wacc[wave][4 * lane + 3] = acc.w;
  if (lane == 0) { wm[wave] = m; wsum[wave] = s; }
  __syncthreads();

  float M = wm[0];
#pragma unroll
  for (int w = 1; w < WPB; ++w) M = fmaxf(M, wm[w]);

  float* bp = part + (size_t)blockIdx.x * PSTRIDE;
  if (tid < NHID) {
    float a = 0.f;
#pragma unroll
    for (int w = 0; w < WPB; ++w) a += wacc[w][tid] * expf(wm[w] - M);
    bp[2 + tid] = a;
  }
  if (tid == 0) {
    float S = 0.f;
#pragma unroll
    for (int w = 0; w < WPB; ++w) S += wsum[w] * expf(wm[w] - M);
    bp[0] = M;
    bp[1] = S;
  }
}

__global__ __launch_bounds__(256)
void sa_finalize(const float* __restrict__ part, float* __restrict__ out) {
  __shared__ float red[256];
  const int tid = threadIdx.x;

  // global max over block partials
  float m = -INFINITY;
  for (int b = tid; b < NB; b += 256) m = fmaxf(m, part[(size_t)b * PSTRIDE]);
  red[tid] = m; __syncthreads();
  for (int o = 128; o > 0; o >>= 1) {
    if (tid < o) red[tid] = fmaxf(red[tid], red[tid + o]);
    __syncthreads();
  }
  const float M = red[0];
  __syncthreads();

  // global exp-sum
  float S = 0.f;
  for (int b = tid; b < NB; b += 256)
    S += part[(size_t)b * PSTRIDE + 1] * expf(part[(size_t)b * PSTRIDE] - M);
  red[tid] = S; __syncthreads();
  for (int o = 128; o > 0; o >>= 1) {
    if (tid < o) red[tid] += red[tid + o];
    __syncthreads();
  }
  S = red[0];

  if (tid < NHID) {
    float num = 0.f;
    for (int b = 0; b < NB; ++b)
      num += part[(size_t)b * PSTRIDE + 2 + tid] * expf(part[(size_t)b * PSTRIDE] - M);
    out[tid] = num / S;
  }
}

extern "C" void kernel_launch(void* const* d_in, const int* in_sizes, int n_in,
                              void* d_out, int out_size, void* d_ws, size_t ws_size,
                              hipStream_t stream) {
  const float* inputs = (const float*)d_in[0];
  const float* W      = (const float*)d_in[1];
  // d_in[2] (b) and d_in[3] (index): own_score + b is identical for every row,
  // and softmax(x + const) == softmax(x), so they cancel analytically.
  float* out  = (float*)d_out;
  float* part = (float*)d_ws;   // NB * 130 floats = ~520 KB

  sa_partial <<<NB, BT, 0, stream>>>(inputs, W, part);
  sa_finalize<<<1, 256, 0, stream>>>(part, out);

  (void)in_sizes; (void)n_in; (void)out_size; (void)ws_size;
}